// TransMIL_per_modality_85280870630068
// MI455X (gfx1250) — compile-verified
//
#include <hip/hip_runtime.h>

// ---------------------------------------------------------------------------
// TransMIL (Nystrom attention x2 + PPEG) for MI455X / gfx1250, fp32 via
// V_WMMA_F32_16X16X4_F32. All GEMMs run on the matrix pipe (wave32 WMMA).
// Workspace use: ~581 MB (aliased; see layout below).
// ---------------------------------------------------------------------------

#define BATCH 4
#define NIN   6170
#define LIN   768
#define DM    512
#define NH    8
#define DH    64
#define NTOK  6264      // 1 + 10 + 169 + 6084
#define NPAD  6400      // NTOK padded (front) to multiple of 256
#define PADF  136       // front pad
#define LM    256       // landmarks
#define LSEG  25        // NPAD / LM
#define NBH   (BATCH*NH)

typedef float v2f __attribute__((ext_vector_type(2)));
typedef float v8f __attribute__((ext_vector_type(8)));

#define WMMA_F32(a, b, c) __builtin_amdgcn_wmma_f32_16x16x4_f32( \
    false, (a), false, (b), (short)0, (c), false, false)

// ---------------------------------------------------------------------------
// Generic batched WMMA f32 GEMM.  C = epil(A @ B)
//   A: MxK row-major (lda).  B: KxN row-major (ldb), or NxK row-major used
//   transposed when BT.  Batch via blockIdx.z with element strides sA/sB/sC.
//   EPIL: 0 -> alpha*(A@B)   1 -> relu(A@B + bias[n])   2 -> A@B + bias[n]
// Block = 64 threads = 2 waves stacked in M; block tile 64x64.  Each wave
// computes a 32x64 tile as 2x4 WMMA 16x16 tiles (A-frags reused across 4
// B-columns -> 8 WMMAs per 4-deep K-step vs 6 fragment loads).
// All pointers/fragments are NAMED SCALARS (no arrays) so clang's address-
// space inference keeps the loads on the GLOBAL path (not FLAT).
// Fragment layout per CDNA5 ISA:
//   A 16x4:  lanes 0-15 -> M=l16, K={k0,k0+1}; lanes 16-31 -> K={k0+2,k0+3}
//   B 4x16:  lanes 0-15 -> N=l16, K={k0,k0+1}; lanes 16-31 -> K={k0+2,k0+3}
//   C/D v8f: element r -> row = 8*half + r, col = l16
// Out-of-range fragment rows/cols are CLAMPED (not zeroed): they only feed
// accumulator elements that the exactly-guarded store never writes, so the
// inner loop stays completely branch-free.
// ---------------------------------------------------------------------------
template<int EPIL>
__device__ __forceinline__ void store_tile(float* C, const v8f& acc, int mb, int nc,
                                           int M, int N, int ldc, int half,
                                           const float* bias, float alpha) {
  if (nc >= N) return;
#pragma unroll
  for (int r = 0; r < 8; ++r) {
    int row = mb + half * 8 + r;
    if (row < M) {
      float v = acc[r];
      if (EPIL == 0) v *= alpha;
      else { v += bias[nc]; if (EPIL == 1) v = fmaxf(v, 0.f); }
      C[(long)row * ldc + nc] = v;
    }
  }
}

template<bool BT, int EPIL>
__global__ __launch_bounds__(64) void gemm_f32_wmma(
    const float* __restrict__ Ab, const float* __restrict__ Bb,
    float* __restrict__ Cb, const float* __restrict__ bias,
    int M, int N, int K, int lda, int ldb, int ldc,
    long sA, long sB, long sC, float alpha)
{
  const float* A = Ab + (long)blockIdx.z * sA;
  const float* B = Bb + (long)blockIdx.z * sB;
  float*       C = Cb + (long)blockIdx.z * sC;

  const int lane = threadIdx.x & 31;
  const int wv   = threadIdx.x >> 5;        // 0..1 (row of wave within block)
  const int half = lane >> 4;
  const int l16  = lane & 15;
  const int m0 = blockIdx.y * 64 + wv * 32; // wave covers rows [m0, m0+32)
  const int n0 = blockIdx.x * 64;           // wave covers cols [n0, n0+64)

  // Clamped fragment source rows / cols (loop-invariant, branch-free body)
  int ra0 = m0 + l16;      if (ra0 >= M) ra0 = M - 1;
  int ra1 = m0 + 16 + l16; if (ra1 >= M) ra1 = M - 1;
  int cb0 = n0 + l16;      if (cb0 >= N) cb0 = N - 1;
  int cb1 = n0 + 16 + l16; if (cb1 >= N) cb1 = N - 1;
  int cb2 = n0 + 32 + l16; if (cb2 >= N) cb2 = N - 1;
  int cb3 = n0 + 48 + l16; if (cb3 >= N) cb3 = N - 1;

  const float* pa0 = A + (long)ra0 * lda + 2 * half;
  const float* pa1 = A + (long)ra1 * lda + 2 * half;
  const float *pb0, *pb1, *pb2, *pb3;
  long stepB;
  if (BT) {
    pb0 = B + (long)cb0 * ldb + 2 * half;
    pb1 = B + (long)cb1 * ldb + 2 * half;
    pb2 = B + (long)cb2 * ldb + 2 * half;
    pb3 = B + (long)cb3 * ldb + 2 * half;
    stepB = 4;
  } else {
    pb0 = B + (long)(2 * half) * ldb + cb0;
    pb1 = B + (long)(2 * half) * ldb + cb1;
    pb2 = B + (long)(2 * half) * ldb + cb2;
    pb3 = B + (long)(2 * half) * ldb + cb3;
    stepB = 4 * (long)ldb;
  }

  v8f c00 = {}, c01 = {}, c02 = {}, c03 = {};
  v8f c10 = {}, c11 = {}, c12 = {}, c13 = {};

#pragma unroll 2
  for (int k0 = 0; k0 < K; k0 += 4) {
    v2f a0 = *(const v2f*)pa0;      // 8-byte loads (ka even, lda even)
    v2f a1 = *(const v2f*)pa1;
    v2f b0, b1, b2, b3;
    if (BT) {
      b0 = *(const v2f*)pb0;
      b1 = *(const v2f*)pb1;
      b2 = *(const v2f*)pb2;
      b3 = *(const v2f*)pb3;
    } else {
      b0.x = pb0[0]; b0.y = pb0[ldb];
      b1.x = pb1[0]; b1.y = pb1[ldb];
      b2.x = pb2[0]; b2.y = pb2[ldb];
      b3.x = pb3[0]; b3.y = pb3[ldb];
    }
    pa0 += 4; pa1 += 4;
    pb0 += stepB; pb1 += stepB; pb2 += stepB; pb3 += stepB;
    c00 = WMMA_F32(a0, b0, c00);
    c01 = WMMA_F32(a0, b1, c01);
    c02 = WMMA_F32(a0, b2, c02);
    c03 = WMMA_F32(a0, b3, c03);
    c10 = WMMA_F32(a1, b0, c10);
    c11 = WMMA_F32(a1, b1, c11);
    c12 = WMMA_F32(a1, b2, c12);
    c13 = WMMA_F32(a1, b3, c13);
  }

  store_tile<EPIL>(C, c00, m0,      n0 +      l16, M, N, ldc, half, bias, alpha);
  store_tile<EPIL>(C, c01, m0,      n0 + 16 + l16, M, N, ldc, half, bias, alpha);
  store_tile<EPIL>(C, c02, m0,      n0 + 32 + l16, M, N, ldc, half, bias, alpha);
  store_tile<EPIL>(C, c03, m0,      n0 + 48 + l16, M, N, ldc, half, bias, alpha);
  store_tile<EPIL>(C, c10, m0 + 16, n0 +      l16, M, N, ldc, half, bias, alpha);
  store_tile<EPIL>(C, c11, m0 + 16, n0 + 16 + l16, M, N, ldc, half, bias, alpha);
  store_tile<EPIL>(C, c12, m0 + 16, n0 + 32 + l16, M, N, ldc, half, bias, alpha);
  store_tile<EPIL>(C, c13, m0 + 16, n0 + 48 + l16, M, N, ldc, half, bias, alpha);
}

// ---------------------------------------------------------------------------
// Elementwise / reduction kernels
// ---------------------------------------------------------------------------

// Assemble H (B,NTOK,DM) from fc1 output (B,NIN,DM) + cls token.
__global__ void build_h(const float* __restrict__ hfc, const float* __restrict__ cls,
                        float* __restrict__ H) {
  long i = (long)blockIdx.x * 256 + threadIdx.x;
  if (i >= (long)BATCH * NTOK * DM) return;
  int c = (int)(i % DM);
  long bn = i / DM;
  int bb = (int)(bn / NTOK), t = (int)(bn % NTOK);
  float v;
  if (t == 0) v = cls[c];
  else if (t < 11) v = hfc[((long)bb * NIN + (t - 1)) * DM + c];
  else if (t < 11 + 169) {
    int p = t - 11; int src = 10 + (p < 160 ? p : p - 160);
    v = hfc[((long)bb * NIN + src) * DM + c];
  } else {
    int p = t - 180; int src = 170 + (p < 6000 ? p : p - 6000);
    v = hfc[((long)bb * NIN + src) * DM + c];
  }
  H[i] = v;
}

// LayerNorm over DM, write into front-zero-padded XP (B,NPAD,DM).
__global__ __launch_bounds__(256) void layernorm_pad(
    const float* __restrict__ H, const float* __restrict__ g,
    const float* __restrict__ b, float* __restrict__ XP) {
  const int idx = blockIdx.x;
  const int bb = idx / NPAD, t = idx % NPAD;
  float* o = XP + (long)idx * DM;
  const int tid = threadIdx.x;
  if (t < PADF) { for (int c = tid; c < DM; c += 256) o[c] = 0.f; return; }
  const float* xr = H + ((long)bb * NTOK + (t - PADF)) * DM;
  __shared__ float red[256];
  float s = 0.f;
  for (int c = tid; c < DM; c += 256) s += xr[c];
  red[tid] = s; __syncthreads();
  for (int st = 128; st > 0; st >>= 1) { if (tid < st) red[tid] += red[tid + st]; __syncthreads(); }
  const float mu = red[0] / DM; __syncthreads();
  float v = 0.f;
  for (int c = tid; c < DM; c += 256) { float d = xr[c] - mu; v += d * d; }
  red[tid] = v; __syncthreads();
  for (int st = 128; st > 0; st >>= 1) { if (tid < st) red[tid] += red[tid + st]; __syncthreads(); }
  const float rstd = rsqrtf(red[0] / DM + 1e-5f);
  for (int c = tid; c < DM; c += 256) o[c] = (xr[c] - mu) * rstd * g[c] + b[c];
}

// Split fused qkv (B,NPAD,3*DM) into Q(scaled)/K/V laid out (B,NH,NPAD,DH).
__global__ void split_qkv(const float* __restrict__ qkv, float* __restrict__ Q,
                          float* __restrict__ K, float* __restrict__ V) {
  long i = (long)blockIdx.x * 256 + threadIdx.x;
  if (i >= (long)BATCH * NPAD * DM) return;
  int c = (int)(i % DM);
  long bn = i / DM;
  int bb = (int)(bn / NPAD), n = (int)(bn % NPAD);
  int h = c >> 6, d = c & 63;
  long src = bn * (3 * DM);
  long dst = (((long)bb * NH + h) * NPAD + n) * DH + d;
  Q[dst] = qkv[src + c] * 0.125f;           // SCALE = DIM_HEAD^-0.5
  K[dst] = qkv[src + DM + c];
  V[dst] = qkv[src + 2 * DM + c];
}

// Landmark means: (B,NH,LM,DH) = mean over LSEG consecutive tokens.
__global__ void landmark_mean(const float* __restrict__ X, float* __restrict__ XL) {
  long i = (long)blockIdx.x * 256 + threadIdx.x;
  if (i >= (long)NBH * LM * DH) return;
  int d = (int)(i & 63);
  long r = i >> 6;
  int m = (int)(r % LM);
  long zh = r / LM;
  const float* src = X + (zh * NPAD + (long)m * LSEG) * DH + d;
  float s = 0.f;
#pragma unroll
  for (int j = 0; j < LSEG; ++j) s += src[(long)j * DH];
  XL[i] = s * (1.f / LSEG);
}

// Row softmax in place; one 256-thread block per row.
__global__ __launch_bounds__(256) void softmax_rows(float* __restrict__ p, int ncols) {
  float* x = p + (long)blockIdx.x * ncols;
  __shared__ float red[256];
  int t = threadIdx.x;
  float m = -3.0e38f;
  for (int c = t; c < ncols; c += 256) m = fmaxf(m, x[c]);
  red[t] = m; __syncthreads();
  for (int s = 128; s > 0; s >>= 1) { if (t < s) red[t] = fmaxf(red[t], red[t + s]); __syncthreads(); }
  m = red[0]; __syncthreads();
  float sum = 0.f;
  for (int c = t; c < ncols; c += 256) { float e = expf(x[c] - m); x[c] = e; sum += e; }
  red[t] = sum; __syncthreads();
  for (int s = 128; s > 0; s >>= 1) { if (t < s) red[t] += red[t + s]; __syncthreads(); }
  float inv = 1.f / red[0];
  for (int c = t; c < ncols; c += 256) x[c] *= inv;
}

// Per-(b,h) max row-sum / max col-sum of |A2| -> norms[2z],norms[2z+1]
__global__ __launch_bounds__(256) void pinv_norms(const float* __restrict__ A,
                                                  float* __restrict__ norms) {
  const float* a = A + (long)blockIdx.x * LM * LM;
  int t = threadIdx.x;
  float rs = 0.f, cs = 0.f;
  for (int j = 0; j < LM; ++j) { rs += fabsf(a[(long)t * LM + j]); cs += fabsf(a[(long)j * LM + t]); }
  __shared__ float r1[256], r2[256];
  r1[t] = rs; r2[t] = cs; __syncthreads();
  for (int s = 128; s > 0; s >>= 1) {
    if (t < s) { r1[t] = fmaxf(r1[t], r1[t + s]); r2[t] = fmaxf(r2[t], r2[t + s]); }
    __syncthreads();
  }
  if (t == 0) { norms[2 * blockIdx.x] = r1[0]; norms[2 * blockIdx.x + 1] = r2[0]; }
}

// Global (across all b,h — matches jnp .max()) scalar 1/(col*row).
__global__ void pinv_scalar(const float* __restrict__ norms, float* __restrict__ scal) {
  if (threadIdx.x == 0) {
    float cm = norms[0], rm = norms[1];
    for (int i = 1; i < NBH; ++i) { cm = fmaxf(cm, norms[2 * i]); rm = fmaxf(rm, norms[2 * i + 1]); }
    scal[0] = 1.f / (cm * rm);
  }
}

// Z = A^T * scal  (batched)
__global__ void transpose_scale(const float* __restrict__ A, float* __restrict__ Z,
                                const float* __restrict__ scal) {
  long i = (long)blockIdx.x * 256 + threadIdx.x;
  if (i >= (long)NBH * LM * LM) return;
  long zb = i / ((long)LM * LM);
  long w = i % ((long)LM * LM);
  int ri = (int)(w / LM), ci = (int)(w % LM);
  Z[i] = A[zb * (long)LM * LM + (long)ci * LM + ri] * scal[0];
}

// T = cv*I - U  (batched)
__global__ void diag_sub(const float* __restrict__ U, float* __restrict__ T, float cv) {
  long i = (long)blockIdx.x * 256 + threadIdx.x;
  if (i >= (long)NBH * LM * LM) return;
  long w = i % ((long)LM * LM);
  T[i] = (((int)(w / LM)) == ((int)(w % LM)) ? cv : 0.f) - U[i];
}

// OH += depthwise conv1d(V) along N, kernel 33, pad 16, per-head weights.
__global__ void dwconv_add(float* __restrict__ OH, const float* __restrict__ V,
                           const float* __restrict__ rw) {
  long i = (long)blockIdx.x * 256 + threadIdx.x;
  if (i >= (long)NBH * NPAD * DH) return;
  int d = (int)(i & 63);
  long r = i >> 6;
  int n = (int)(r % NPAD);
  long zh = r / NPAD;
  int h = (int)(zh % NH);
  const float* vb = V + zh * (long)NPAD * DH + d;
  float s = 0.f;
#pragma unroll
  for (int t = 0; t < 33; ++t) {
    int nn = n + t - 16;
    if (nn >= 0 && nn < NPAD) s += vb[(long)nn * DH] * rw[h * 33 + t];
  }
  OH[i] += s;
}

// (B,NH,NPAD,DH) -> (B,NPAD,DM)
__global__ void merge_heads(const float* __restrict__ OH, float* __restrict__ MG) {
  long i = (long)blockIdx.x * 256 + threadIdx.x;
  if (i >= (long)BATCH * NPAD * DM) return;
  int c = (int)(i % DM);
  long bn = i / DM;
  int bb = (int)(bn / NPAD), n = (int)(bn % NPAD);
  int h = c >> 6, d = c & 63;
  MG[i] = OH[(((long)bb * NH + h) * NPAD + n) * DH + d];
}

// H += Y[:, -NTOK:, :]  (drop the front pad)
__global__ void add_out(float* __restrict__ H, const float* __restrict__ Y) {
  long i = (long)blockIdx.x * 256 + threadIdx.x;
  if (i >= (long)BATCH * NTOK * DM) return;
  int c = (int)(i % DM);
  long bn = i / DM;
  int bb = (int)(bn / NTOK), n = (int)(bn % NTOK);
  H[i] += Y[((long)bb * NPAD + n + PADF) * DM + c];
}

// PPEG: copy tokens [0,11) unchanged.
__global__ void ppeg_copyhead(const float* __restrict__ H, float* __restrict__ HN) {
  long i = (long)blockIdx.x * 256 + threadIdx.x;
  if (i >= (long)BATCH * 11 * DM) return;
  int c = (int)(i % DM);
  long bn = i / DM;
  int bb = (int)(bn / 11), t = (int)(bn % 11);
  HN[((long)bb * NTOK + t) * DM + c] = H[((long)bb * NTOK + t) * DM + c];
}

// PPEG conv block: y = x + dw7(x)+b7 + dw5(x)+b5 + dw3(x)+b3 on an SxS grid of
// tokens starting at `base`, channels-last in H.
__global__ void ppeg_conv(const float* __restrict__ H, float* __restrict__ HN,
                          int base, int S,
                          const float* __restrict__ w7, const float* __restrict__ b7,
                          const float* __restrict__ w5, const float* __restrict__ b5,
                          const float* __restrict__ w3, const float* __restrict__ b3) {
  long i = (long)blockIdx.x * 256 + threadIdx.x;
  long total = (long)BATCH * S * S * DM;
  if (i >= total) return;
  int c = (int)(i % DM);
  long r = i / DM;
  int j = (int)(r % S); r /= S;
  int ii = (int)(r % S);
  int bb = (int)(r / S);
  const float* hb = H + (long)bb * NTOK * DM;
  float a7 = b7[c], a5 = b5[c], a3 = b3[c];
  for (int dy = 0; dy < 7; ++dy)
    for (int dx = 0; dx < 7; ++dx) {
      int y = ii + dy - 3, x = j + dx - 3;
      if (y >= 0 && y < S && x >= 0 && x < S)
        a7 += hb[(long)(base + y * S + x) * DM + c] * w7[c * 49 + dy * 7 + dx];
    }
  for (int dy = 0; dy < 5; ++dy)
    for (int dx = 0; dx < 5; ++dx) {
      int y = ii + dy - 2, x = j + dx - 2;
      if (y >= 0 && y < S && x >= 0 && x < S)
        a5 += hb[(long)(base + y * S + x) * DM + c] * w5[c * 25 + dy * 5 + dx];
    }
  for (int dy = 0; dy < 3; ++dy)
    for (int dx = 0; dx < 3; ++dx) {
      int y = ii + dy - 1, x = j + dx - 1;
      if (y >= 0 && y < S && x >= 0 && x < S)
        a3 += hb[(long)(base + y * S + x) * DM + c] * w3[c * 9 + dy * 3 + dx];
    }
  float xv = hb[(long)(base + ii * S + j) * DM + c];
  HN[((long)bb * NTOK + base + ii * S + j) * DM + c] = xv + a7 + a5 + a3;
}

// Final layernorm of token 0 only -> out (B,DM)
__global__ __launch_bounds__(256) void final_norm(const float* __restrict__ H,
    const float* __restrict__ g, const float* __restrict__ b, float* __restrict__ out) {
  const int bb = blockIdx.x;
  const float* xr = H + (long)bb * NTOK * DM;
  __shared__ float red[256];
  int tid = threadIdx.x;
  float s = 0.f;
  for (int c = tid; c < DM; c += 256) s += xr[c];
  red[tid] = s; __syncthreads();
  for (int st = 128; st > 0; st >>= 1) { if (tid < st) red[tid] += red[tid + st]; __syncthreads(); }
  float mu = red[0] / DM; __syncthreads();
  float v = 0.f;
  for (int c = tid; c < DM; c += 256) { float d = xr[c] - mu; v += d * d; }
  red[tid] = v; __syncthreads();
  for (int st = 128; st > 0; st >>= 1) { if (tid < st) red[tid] += red[tid + st]; __syncthreads(); }
  float rstd = rsqrtf(red[0] / DM + 1e-5f);
  for (int c = tid; c < DM; c += 256) out[(long)bb * DM + c] = (xr[c] - mu) * rstd * g[c] + b[c];
}

// ---------------------------------------------------------------------------
// Host orchestration
// ---------------------------------------------------------------------------
namespace {

struct Bufs {
  float *H, *H2, *XP, *R1, *Q, *K, *V, *QL, *KL;
  float *A2, *Z, *Z2, *AZ, *T, *U, *KV, *ZKV, *NRM;
};

inline dim3 g1(long total) { return dim3((unsigned)((total + 255) / 256)); }

void run_attention(hipStream_t s, float* H, const float* ng, const float* nb,
                   const float* wqkv, const float* wout, const float* bout,
                   const float* resw, const Bufs& bf) {
  const long sHV = (long)NPAD * DH;   // per-(b,h) stride for Q/K/V
  const long sL  = (long)LM * DH;     // per-(b,h) stride for QL/KL/KV/ZKV
  const long sS  = (long)LM * LM;     // per-(b,h) stride for 256x256 mats

  layernorm_pad<<<BATCH * NPAD, 256, 0, s>>>(H, ng, nb, bf.XP);

  { // fused qkv GEMM: (B*NPAD,512) @ (512,1536)
    dim3 g((3 * DM + 63) / 64, (BATCH * NPAD + 63) / 64, 1);
    gemm_f32_wmma<false, 0><<<g, 64, 0, s>>>(bf.XP, wqkv, bf.R1, nullptr,
        BATCH * NPAD, 3 * DM, DM, DM, 3 * DM, 3 * DM, 0, 0, 0, 1.f);
  }
  split_qkv<<<g1((long)BATCH * NPAD * DM), 256, 0, s>>>(bf.R1, bf.Q, bf.K, bf.V);
  landmark_mean<<<g1((long)NBH * LM * DH), 256, 0, s>>>(bf.Q, bf.QL);
  landmark_mean<<<g1((long)NBH * LM * DH), 256, 0, s>>>(bf.K, bf.KL);

  { // sim3 = QL @ K^T : (LM x NPAD), batched 32, into R1
    dim3 g((NPAD + 63) / 64, (LM + 63) / 64, NBH);
    gemm_f32_wmma<true, 0><<<g, 64, 0, s>>>(bf.QL, bf.K, bf.R1, nullptr,
        LM, NPAD, DH, DH, DH, NPAD, sL, sHV, (long)LM * NPAD, 1.f);
  }
  softmax_rows<<<NBH * LM, 256, 0, s>>>(bf.R1, NPAD);
  { // KV = a3 @ V : (LM x DH), K=NPAD
    dim3 g(1, (LM + 63) / 64, NBH);
    gemm_f32_wmma<false, 0><<<g, 64, 0, s>>>(bf.R1, bf.V, bf.KV, nullptr,
        LM, DH, NPAD, NPAD, DH, DH, (long)LM * NPAD, sHV, sL, 1.f);
  }
  { // a2 = QL @ KL^T : (LM x LM)
    dim3 g((LM + 63) / 64, (LM + 63) / 64, NBH);
    gemm_f32_wmma<true, 0><<<g, 64, 0, s>>>(bf.QL, bf.KL, bf.A2, nullptr,
        LM, LM, DH, DH, DH, LM, sL, sL, sS, 1.f);
  }
  softmax_rows<<<NBH * LM, 256, 0, s>>>(bf.A2, LM);
  { // a1 = Q @ KL^T : (NPAD x LM), into R1 (sim3 no longer needed)
    dim3 g((LM + 63) / 64, (NPAD + 63) / 64, NBH);
    gemm_f32_wmma<true, 0><<<g, 64, 0, s>>>(bf.Q, bf.KL, bf.R1, nullptr,
        NPAD, LM, DH, DH, DH, LM, sHV, sL, (long)NPAD * LM, 1.f);
  }
  softmax_rows<<<NBH * NPAD, 256, 0, s>>>(bf.R1, LM);

  // Newton-Schulz pseudo-inverse of a2 (6 iterations)
  pinv_norms<<<NBH, 256, 0, s>>>(bf.A2, bf.NRM);
  pinv_scalar<<<1, 32, 0, s>>>(bf.NRM, bf.NRM + 64);
  transpose_scale<<<g1((long)NBH * sS), 256, 0, s>>>(bf.A2, bf.Z, bf.NRM + 64);
  float* zc = bf.Z; float* zn = bf.Z2;
  dim3 gS((LM + 63) / 64, (LM + 63) / 64, NBH);
  for (int it = 0; it < 6; ++it) {
    gemm_f32_wmma<false, 0><<<gS, 64, 0, s>>>(bf.A2, zc, bf.AZ, nullptr,
        LM, LM, LM, LM, LM, LM, sS, sS, sS, 1.f);
    diag_sub<<<g1((long)NBH * sS), 256, 0, s>>>(bf.AZ, bf.T, 7.f);
    gemm_f32_wmma<false, 0><<<gS, 64, 0, s>>>(bf.AZ, bf.T, bf.U, nullptr,
        LM, LM, LM, LM, LM, LM, sS, sS, sS, 1.f);
    diag_sub<<<g1((long)NBH * sS), 256, 0, s>>>(bf.U, bf.T, 15.f);
    gemm_f32_wmma<false, 0><<<gS, 64, 0, s>>>(bf.AZ, bf.T, bf.U, nullptr,
        LM, LM, LM, LM, LM, LM, sS, sS, sS, 1.f);
    diag_sub<<<g1((long)NBH * sS), 256, 0, s>>>(bf.U, bf.T, 13.f);
    gemm_f32_wmma<false, 0><<<gS, 64, 0, s>>>(zc, bf.T, zn, nullptr,
        LM, LM, LM, LM, LM, LM, sS, sS, sS, 0.25f);
    float* tmp = zc; zc = zn; zn = tmp;
  }
  { // ZKV = Z @ KV : (LM x DH)
    dim3 g(1, (LM + 63) / 64, NBH);
    gemm_f32_wmma<false, 0><<<g, 64, 0, s>>>(zc, bf.KV, bf.ZKV, nullptr,
        LM, DH, LM, LM, DH, DH, sS, sL, sL, 1.f);
  }
  { // OH = a1 @ ZKV : (NPAD x DH), into XP region (free after qkv GEMM)
    dim3 g(1, (NPAD + 63) / 64, NBH);
    gemm_f32_wmma<false, 0><<<g, 64, 0, s>>>(bf.R1, bf.ZKV, bf.XP, nullptr,
        NPAD, DH, LM, LM, DH, DH, (long)NPAD * LM, sL, sHV, 1.f);
  }
  dwconv_add<<<g1((long)NBH * NPAD * DH), 256, 0, s>>>(bf.XP, bf.V, resw);
  merge_heads<<<g1((long)BATCH * NPAD * DM), 256, 0, s>>>(bf.XP, bf.K);
  { // Y = MG @ wout + bout : (B*NPAD x DM), into Q region
    dim3 g((DM + 63) / 64, (BATCH * NPAD + 63) / 64, 1);
    gemm_f32_wmma<false, 2><<<g, 64, 0, s>>>(bf.K, wout, bf.Q, bout,
        BATCH * NPAD, DM, DM, DM, DM, DM, 0, 0, 0, 1.f);
  }
  add_out<<<g1((long)BATCH * NTOK * DM), 256, 0, s>>>(H, bf.Q);
}

} // namespace

extern "C" void kernel_launch(void* const* d_in, const int* in_sizes, int n_in,
                              void* d_out, int out_size, void* d_ws, size_t ws_size,
                              hipStream_t stream) {
  (void)in_sizes; (void)n_in; (void)out_size; (void)ws_size;
  const float* x       = (const float*)d_in[0];
  const float* fc1_w   = (const float*)d_in[1];
  const float* fc1_b   = (const float*)d_in[2];
  const float* cls     = (const float*)d_in[3];
  const float* l1_ng   = (const float*)d_in[4];
  const float* l1_nb   = (const float*)d_in[5];
  const float* l1_qkv  = (const float*)d_in[6];
  const float* l1_ow   = (const float*)d_in[7];
  const float* l1_ob   = (const float*)d_in[8];
  const float* l1_res  = (const float*)d_in[9];
  const float* ct_w7   = (const float*)d_in[10];
  const float* ct_b7   = (const float*)d_in[11];
  const float* ct_w5   = (const float*)d_in[12];
  const float* ct_b5   = (const float*)d_in[13];
  const float* ct_w3   = (const float*)d_in[14];
  const float* ct_b3   = (const float*)d_in[15];
  const float* pt_w7   = (const float*)d_in[16];
  const float* pt_b7   = (const float*)d_in[17];
  const float* pt_w5   = (const float*)d_in[18];
  const float* pt_b5   = (const float*)d_in[19];
  const float* pt_w3   = (const float*)d_in[20];
  const float* pt_b3   = (const float*)d_in[21];
  const float* l2_ng   = (const float*)d_in[22];
  const float* l2_nb   = (const float*)d_in[23];
  const float* l2_qkv  = (const float*)d_in[24];
  const float* l2_ow   = (const float*)d_in[25];
  const float* l2_ob   = (const float*)d_in[26];
  const float* l2_res  = (const float*)d_in[27];
  const float* norm_g  = (const float*)d_in[28];
  const float* norm_b  = (const float*)d_in[29];
  float* out = (float*)d_out;

  // Workspace layout (floats). Total ~145.2M floats = ~581 MB.
  float* W = (float*)d_ws;
  size_t o = 0;
  const size_t szH  = (size_t)BATCH * NTOK * DM;       // 12,828,672
  const size_t szXP = (size_t)BATCH * NPAD * DM;       // 13,107,200 (HFC -> XP -> OH)
  const size_t szR1 = (size_t)NBH * NPAD * LM;         // 52,428,800 (QKV -> SIM)
  const size_t szHV = (size_t)NBH * NPAD * DH;         // 13,107,200
  const size_t szL  = (size_t)NBH * LM * DH;           //    524,288
  const size_t szS  = (size_t)NBH * LM * LM;           //  2,097,152
  Bufs bf;
  bf.H   = W + o; o += szH;
  bf.H2  = W + o; o += szH;
  bf.XP  = W + o; o += szXP;
  bf.R1  = W + o; o += szR1;
  bf.Q   = W + o; o += szHV;   // also Y
  bf.K   = W + o; o += szHV;   // also MG
  bf.V   = W + o; o += szHV;
  bf.QL  = W + o; o += szL;
  bf.KL  = W + o; o += szL;
  bf.A2  = W + o; o += szS;
  bf.Z   = W + o; o += szS;
  bf.Z2  = W + o; o += szS;
  bf.AZ  = W + o; o += szS;
  bf.T   = W + o; o += szS;
  bf.U   = W + o; o += szS;
  bf.KV  = W + o; o += szL;
  bf.ZKV = W + o; o += szL;
  bf.NRM = W + o; o += 128;    // [0..63] per-bh norms, [64] global scalar

  // 1) fc1 + relu : (B*NIN,768)@(768,512) -> HFC (in XP region)
  {
    dim3 g((DM + 63) / 64, (BATCH * NIN + 63) / 64, 1);
    gemm_f32_wmma<false, 1><<<g, 64, 0, stream>>>(x, fc1_w, bf.XP, fc1_b,
        BATCH * NIN, DM, LIN, LIN, DM, DM, 0, 0, 0, 1.f);
  }
  // 2) assemble tokens (cls + wrap-padded ct/pth blocks)
  build_h<<<g1((long)BATCH * NTOK * DM), 256, 0, stream>>>(bf.XP, cls, bf.H);

  // 3) attention layer 1 (in-place on H)
  run_attention(stream, bf.H, l1_ng, l1_nb, l1_qkv, l1_ow, l1_ob, l1_res, bf);

  // 4) PPEG: H -> H2 (tokens 0..10 copied, conv blocks rewritten)
  ppeg_copyhead<<<g1((long)BATCH * 11 * DM), 256, 0, stream>>>(bf.H, bf.H2);
  ppeg_conv<<<g1((long)BATCH * 13 * 13 * DM), 256, 0, stream>>>(bf.H, bf.H2, 11, 13,
      ct_w7, ct_b7, ct_w5, ct_b5, ct_w3, ct_b3);
  ppeg_conv<<<g1((long)BATCH * 78 * 78 * DM), 256, 0, stream>>>(bf.H, bf.H2, 180, 78,
      pt_w7, pt_b7, pt_w5, pt_b5, pt_w3, pt_b3);

  // 5) attention layer 2 (in-place on H2)
  run_attention(stream, bf.H2, l2_ng, l2_nb, l2_qkv, l2_ow, l2_ob, l2_res, bf);

  // 6) final layernorm of token 0 -> out (B,DM)
  final_norm<<<BATCH, 256, 0, stream>>>(bf.H2, norm_g, norm_b, out);
}